// DepthAvgPooling_73564199846425
// MI455X (gfx1250) — compile-verified
//
#include <hip/hip_runtime.h>

// Depth-guided 3x3 average pooling, MI455X (gfx1250).
// Pure-bandwidth kernel: ~512 MiB traffic -> ~23us at 23.3 TB/s floor.
// Weights depend only on (b,y,x): computed once per thread, normalized
// (division folded away), OOB taps zeroed (so img loads can clamp coords),
// kept in 9 VGPRs and reused across all channels. Channel planes stream
// through LDS via global_load_async_to_lds_b32 (ASYNCcnt), 3-deep software
// pipeline, 8 LDS buffers -> a single workgroup barrier per channel.

#define BATCH   16
#define CHANS   256
#define HEIGHT  128
#define WIDTH   128
#define ALPHA   8.3f

#define TW      32          // tile width  (one row of threads)
#define THT     8           // tile height
#define TROW    (THT + 2)   // 10 halo rows
#define TCOL    34          // 32 + 2 halo cols (used)
#define TCOLP   36          // padded LDS row stride (floats)
#define TELEMS  (TROW * TCOL)   // 340 elements per tile
#define TBYTES  (TROW * TCOLP * 4)

#define CH_SPLIT    4
#define CH_PER_BLK  (CHANS / CH_SPLIT)   // 64
#define PIPE        3
#define NBUF        8                    // > PIPE+1 -> no trailing barrier

__global__ __launch_bounds__(256)
void depth_pool_kernel(const float* __restrict__ img,
                       const float* __restrict__ depth,
                       float* __restrict__ out)
{
    __shared__ float tile[NBUF][TROW][TCOLP];   // 11520 B of 320 KB

    const int tid = threadIdx.x;
    const int px  = tid & (TW - 1);
    const int py  = tid >> 5;
    const int bx  = blockIdx.x;          // 0..3
    const int by  = blockIdx.y;          // 0..15
    const int bz  = blockIdx.z;          // 0..63
    const int b     = bz >> 2;
    const int chunk = bz & 3;

    const int x   = bx * TW + px;
    const int y   = by * THT + py;
    const int gx0 = bx * TW - 1;
    const int gy0 = by * THT - 1;

    // ---- per-pixel normalized weights, kept in registers for all channels ----
    const float center = depth[((size_t)b * HEIGHT + y) * WIDTH + x];
    float wn[9];
    float den = 0.0f;
#pragma unroll
    for (int di = 0; di < 3; ++di) {
#pragma unroll
        for (int dj = 0; dj < 3; ++dj) {
            const int ny = y + di - 1;
            const int nx = x + dj - 1;
            const bool inb = (ny >= 0) && (ny < HEIGHT) && (nx >= 0) && (nx < WIDTH);
            // reference zero-pads depth: OOB taps use d=0 and still count in den
            const float d = inb ? depth[((size_t)b * HEIGHT + ny) * WIDTH + nx] : 0.0f;
            const float w = __expf(-ALPHA * fabsf(center - d));
            den += w;
            // OOB taps multiply img_pad==0 -> fold the zero into the weight
            wn[di * 3 + dj] = inb ? w : 0.0f;
        }
    }
    const float rden = 1.0f / den;
#pragma unroll
    for (int t = 0; t < 9; ++t) wn[t] *= rden;

    // ---- this thread's two async-load slots (constant across channels) ----
    // img coords clamped into the image (OOB taps have weight 0 -> value unused)
    int e0 = tid;
    int e1 = tid + 256; if (e1 > TELEMS - 1) e1 = TELEMS - 1;
    const int row0 = e0 / TCOL, col0 = e0 % TCOL;
    const int row1 = e1 / TCOL, col1 = e1 % TCOL;
    int gy_0 = gy0 + row0; gy_0 = gy_0 < 0 ? 0 : (gy_0 > HEIGHT - 1 ? HEIGHT - 1 : gy_0);
    int gx_0 = gx0 + col0; gx_0 = gx_0 < 0 ? 0 : (gx_0 > WIDTH  - 1 ? WIDTH  - 1 : gx_0);
    int gy_1 = gy0 + row1; gy_1 = gy_1 < 0 ? 0 : (gy_1 > HEIGHT - 1 ? HEIGHT - 1 : gy_1);
    int gx_1 = gx0 + col1; gx_1 = gx_1 < 0 ? 0 : (gx_1 > WIDTH  - 1 ? WIDTH  - 1 : gx_1);
    const int voff0 = (gy_0 * WIDTH + gx_0) * 4;   // byte offset within channel plane
    const int voff1 = (gy_1 * WIDTH + gx_1) * 4;
    // low 32 bits of a flat shared-memory address == LDS byte address
    const unsigned ldsBase = (unsigned)(unsigned long long)(&tile[0][0][0]);
    const unsigned loff0 = (unsigned)((row0 * TCOLP + col0) * 4);
    const unsigned loff1 = (unsigned)((row1 * TCOLP + col1) * 4);

    const size_t plane = (size_t)HEIGHT * WIDTH;
    const int chBeg = chunk * CH_PER_BLK;
    const float* imgB = img + ((size_t)b * CHANS) * plane;

    auto prefetch = [&](int ch, int buf) {
        const float* src = imgB + (size_t)ch * plane;       // uniform -> SGPR pair
        const unsigned lb = ldsBase + (unsigned)(buf * TBYTES);
        const unsigned l0 = lb + loff0;
        const unsigned l1 = lb + loff1;
        asm volatile("global_load_async_to_lds_b32 %0, %1, %2"
                     :: "v"(l0), "v"(voff0), "s"(src) : "memory");
        asm volatile("global_load_async_to_lds_b32 %0, %1, %2"
                     :: "v"(l1), "v"(voff1), "s"(src) : "memory");
    };

    // prime the pipeline: channels chBeg .. chBeg+2 into buffers 0..2
#pragma unroll
    for (int p = 0; p < PIPE; ++p) prefetch(chBeg + p, p);

    float* outP = out + ((size_t)(b * CHANS + chBeg)) * plane + (size_t)y * WIDTH + x;

    for (int i = 0; i < CH_PER_BLK; ++i) {
        const int pn = i + PIPE;
        if (pn < CH_PER_BLK) {
            prefetch(chBeg + pn, pn & (NBUF - 1));
            // 2 async instrs per prefetch per wave; in-order completion:
            // <=6 outstanding => group i's 2 loads have landed
            asm volatile("s_wait_asynccnt 6" ::: "memory");
        } else {
            const int left = CH_PER_BLK - 1 - i;   // prefetch groups beyond current
            if (left == 2)      asm volatile("s_wait_asynccnt 4" ::: "memory");
            else if (left == 1) asm volatile("s_wait_asynccnt 2" ::: "memory");
            else                asm volatile("s_wait_asynccnt 0" ::: "memory");
        }
        __syncthreads();   // all waves' loads of buffer i visible
                           // this iteration's prefetch, issued before it, writes a buffer 4 (mod 8)
                           // past buffer i-1, which a slower wave may still be reading

        const float* tb = &tile[i & (NBUF - 1)][py][px];
        float acc = 0.0f;
#pragma unroll
        for (int di = 0; di < 3; ++di)
#pragma unroll
            for (int dj = 0; dj < 3; ++dj)
                acc = fmaf(wn[di * 3 + dj], tb[di * TCOLP + dj], acc);

        __builtin_nontemporal_store(acc, outP);
        outP += plane;
    }
}

extern "C" void kernel_launch(void* const* d_in, const int* in_sizes, int n_in,
                              void* d_out, int out_size, void* d_ws, size_t ws_size,
                              hipStream_t stream) {
    const float* img   = (const float*)d_in[0];
    const float* depth = (const float*)d_in[1];
    float* out = (float*)d_out;
    (void)in_sizes; (void)n_in; (void)out_size; (void)d_ws; (void)ws_size;

    dim3 grid(WIDTH / TW, HEIGHT / THT, BATCH * CH_SPLIT);   // 4 x 16 x 64
    depth_pool_kernel<<<grid, 256, 0, stream>>>(img, depth, out);
}